// Counting_Encoding_73650099191998
// MI455X (gfx1250) — compile-verified
//
#include <hip/hip_runtime.h>
#include <stdint.h>

// Segmented histogram, one workgroup per graph segment.
//   out[g,c] = sum of node_weights[i] for i in [ptr[g], ptr[g+1]) with x[i]==c, c < out_dim
//
// Memory-bound problem (~80 MB traffic -> ~3.4us at 23.3 TB/s). Strategy:
//  - per-WG LDS histogram (ds_add_f32 atomics, zero global atomics)
//  - gfx1250 Tensor Data Mover DMA of label chunks into LDS, double buffered,
//    synchronized with s_wait_tensorcnt (TDM ops are in-order per wave)
//  - weights streamed with non-temporal loads
//  - one coalesced 256-float store per graph

#define WG_THREADS 256
#define CHUNK      1024          // staged labels per TDM transfer (4 KB)
#define HIST_MAX   512           // >= out_dim (256)

typedef unsigned int u32x4 __attribute__((ext_vector_type(4)));
typedef int          i32x8 __attribute__((ext_vector_type(8)));
typedef int          i32x4 __attribute__((ext_vector_type(4)));

#if defined(__AMDGCN__) && __has_builtin(__builtin_amdgcn_tensor_load_to_lds) && \
    __has_builtin(__builtin_amdgcn_s_wait_tensorcnt)
#define USE_TDM 1
#else
#define USE_TDM 0
#endif

#if USE_TDM
// Issue a 1-D TENSOR_LOAD_TO_LDS: nelem int32 elements from gsrc -> LDS byte
// address lds_byte_addr. D# layout per CDNA5 ISA ch. 8 (groups 0/1; groups 2/3
// zeroed -> dims 2..4 unused for a 1-D tile). This toolchain (clang-23 /
// amdgpu-toolchain) exposes the 6-arg builtin form.
__device__ __forceinline__ void tdm_load_1d(uint32_t lds_byte_addr,
                                            const int* gsrc, int nelem)
{
  const uint64_t ga = (uint64_t)(uintptr_t)gsrc;
  u32x4 g0;
  g0.x = 1u;                                          // count=1, user descriptor
  g0.y = lds_byte_addr;                               // lds_addr [63:32]
  g0.z = (uint32_t)ga;                                // global_addr [95:64]
  g0.w = (uint32_t)((ga >> 32) & 0x01FFFFFFu)         // global_addr [120:96]
       | (2u << 30);                                  // type=2 ("image")
  i32x8 g1;
  const uint32_t n = (uint32_t)nelem;
  g1.s0 = (int)(2u << 16);                            // data_size=2 (4B); no multicast/pad/iterate
  g1.s1 = (int)((n & 0xFFFFu) << 16);                 // tensor_dim0[15:0] @bits 63:48
  g1.s2 = (int)(((n >> 16) & 0xFFFFu) | (1u << 16));  // tensor_dim0[31:16] | tensor_dim1=1
  g1.s3 = (int)((n & 0xFFFFu) << 16);                 // tile_dim0 = nelem @bits 127:112
  g1.s4 = 0;                                          // tile_dim1/2 unused
  g1.s5 = (int)n;                                     // tensor_dim0_stride (benign for 1 row)
  g1.s6 = 0;
  g1.s7 = 0;
  i32x4 z4 = {0, 0, 0, 0};
  i32x8 z8 = {0, 0, 0, 0, 0, 0, 0, 0};
  __builtin_amdgcn_tensor_load_to_lds(g0, g1, z4, z4, z8, 0);
}
#endif

__global__ __launch_bounds__(WG_THREADS)
void seg_hist_tdm_kernel(const int* __restrict__ x,
                         const int* __restrict__ ptr,
                         const float* __restrict__ w,
                         float* __restrict__ out,
                         int out_dim)
{
  __shared__ float hist[HIST_MAX];
#if USE_TDM
  __shared__ int xbuf[2][CHUNK];
#endif

  const int g     = blockIdx.x;
  const int tid   = threadIdx.x;
  const int start = ptr[g];
  const int end   = ptr[g + 1];

  for (int c = tid; c < out_dim; c += WG_THREADS) hist[c] = 0.0f;

#if USE_TDM
  const int total   = end - start;
  const int nchunks = (total + CHUNK - 1) / CHUNK;
  const bool wave0  = (tid < 32);           // wave32: first wave drives the TDM

  // LDS byte offsets of the staging buffers (low 32 bits of a generic LDS ptr)
  const uint32_t lds_b0 = (uint32_t)(uintptr_t)(&xbuf[0][0]);
  const uint32_t lds_b1 = (uint32_t)(uintptr_t)(&xbuf[1][0]);

  if (nchunks > 0 && wave0) {
    tdm_load_1d(lds_b0, x + start, (total < CHUNK) ? total : CHUNK);
  }

  for (int k = 0; k < nchunks; ++k) {
    const int base = start + k * CHUNK;
    const int rem  = end - base;
    const int n    = (rem < CHUNK) ? rem : CHUNK;

    if (wave0) {
      if (k + 1 < nchunks) {
        const int nbase = base + CHUNK;
        const int nrem  = end - nbase;
        const int nn    = (nrem < CHUNK) ? nrem : CHUNK;
        tdm_load_1d(((k + 1) & 1) ? lds_b1 : lds_b0, x + nbase, nn);
        __builtin_amdgcn_s_wait_tensorcnt(1);   // chunk k complete (in-order)
      } else {
        __builtin_amdgcn_s_wait_tensorcnt(0);   // last chunk complete
      }
    }
    __syncthreads();                            // chunk k visible (and hist zeroed on k==0)

    const int* xs = xbuf[k & 1];
    for (int j = tid; j < n; j += WG_THREADS) {
      const int   label = xs[j];                          // ds_load from staged LDS
      const float wv    = __builtin_nontemporal_load(&w[base + j]);
      if ((unsigned)label < (unsigned)out_dim)
        atomicAdd(&hist[label], wv);                      // ds_add_f32, near-zero contention
    }
    __syncthreads();                            // buffer consumed before reuse at k+2
  }
#else
  __syncthreads();
  for (int i = start + tid; i < end; i += WG_THREADS) {
    const int   label = x[i];
    const float wv    = __builtin_nontemporal_load(&w[i]);
    if ((unsigned)label < (unsigned)out_dim)
      atomicAdd(&hist[label], wv);
  }
#endif

  __syncthreads();
  float* o = out + (size_t)g * (size_t)out_dim;
  for (int c = tid; c < out_dim; c += WG_THREADS) o[c] = hist[c];
}

extern "C" void kernel_launch(void* const* d_in, const int* in_sizes, int n_in,
                              void* d_out, int out_size, void* d_ws, size_t ws_size,
                              hipStream_t stream) {
  (void)n_in; (void)d_ws; (void)ws_size;
  const int*   x   = (const int*)d_in[0];
  const int*   ptr = (const int*)d_in[1];
  const float* w   = (const float*)d_in[2];
  float*       out = (float*)d_out;

  const int num_graphs = in_sizes[1] - 1;          // ptr has num_graphs+1 entries
  const int out_dim    = out_size / num_graphs;    // 256

  seg_hist_tdm_kernel<<<num_graphs, WG_THREADS, 0, stream>>>(x, ptr, w, out, out_dim);
}